// FusedGCNLayer_2714419331421
// MI455X (gfx1250) — compile-verified
//
#include <hip/hip_runtime.h>

#define N_NODES 100000
#define N_EDGES 3200000
#define FEAT    256
#define EMBED   128

typedef __attribute__((ext_vector_type(2))) float v2f;
typedef __attribute__((ext_vector_type(8))) float v8f;
// Exact pointee type from the builtin's diagnostic: vector_size(16) int.
typedef int v4i_b128 __attribute__((vector_size(4 * sizeof(int))));

// 16-row x-tile staged in LDS. Stride 260 floats: 260 % 64 = 4, so the 16 rows
// read by WMMA lanes land on distinct banks (4*row apart) -> no bank conflict.
#define LDS_STRIDE 260
// M-tiles processed per block: 100000 rows = 1250 blocks * 5 tiles * 16 rows.
#define TILES_PER_BLOCK 5

#if defined(__HIP_DEVICE_COMPILE__) && __has_builtin(__builtin_amdgcn_global_load_async_to_lds_b128)
#define HAVE_ASYNC_LDS 1
#else
#define HAVE_ASYNC_LDS 0
#endif

__device__ __forceinline__ void wait_async_lds() {
#if HAVE_ASYNC_LDS
#if __has_builtin(__builtin_amdgcn_s_wait_asynccnt)
    __builtin_amdgcn_s_wait_asynccnt(0);
#else
    asm volatile("s_wait_asynccnt 0" ::: "memory");
#endif
#endif
}

#if HAVE_ASYNC_LDS
__device__ __forceinline__ void async_copy_b128(const float* gsrc, float* ldst, int imm_off4) {
    // Step-wise casts: drop const, add address space, repun pointee to v4i.
    auto g_f = (__attribute__((address_space(1))) const float*)gsrc;
    auto g_v = (__attribute__((address_space(1))) v4i_b128*)g_f;
    auto l_f = (__attribute__((address_space(3))) float*)ldst;
    auto l_v = (__attribute__((address_space(3))) v4i_b128*)l_f;
    switch (imm_off4) {  // keep offset an immediate per the Ii signature
    case 0:  __builtin_amdgcn_global_load_async_to_lds_b128(g_v, l_v,  0, 0); break;
    case 1:  __builtin_amdgcn_global_load_async_to_lds_b128(g_v, l_v, 16, 0); break;
    case 2:  __builtin_amdgcn_global_load_async_to_lds_b128(g_v, l_v, 32, 0); break;
    default: __builtin_amdgcn_global_load_async_to_lds_b128(g_v, l_v, 48, 0); break;
    }
}
#endif

// Stage one 16x256 f32 tile of x into an LDS buffer (256 threads, 16B/lane x4).
__device__ __forceinline__ void stage_tile(const float* __restrict__ x,
                                           int rowbase, float* __restrict__ buf,
                                           int tid) {
    const int row = tid >> 4;             // 0..15
    const int k0  = (tid & 15) << 4;      // 0,16,...,240
    const float* gsrc = x + (size_t)(rowbase + row) * FEAT + k0;
    float*       ldst = buf + row * LDS_STRIDE + k0;
#if HAVE_ASYNC_LDS
    async_copy_b128(gsrc, ldst, 0);
    async_copy_b128(gsrc, ldst, 1);
    async_copy_b128(gsrc, ldst, 2);
    async_copy_b128(gsrc, ldst, 3);
#else
    const float4* s4 = (const float4*)gsrc;
    float4 a0 = s4[0], a1 = s4[1], a2 = s4[2], a3 = s4[3];
    *(float4*)(ldst +  0) = a0;
    *(float4*)(ldst +  4) = a1;
    *(float4*)(ldst +  8) = a2;
    *(float4*)(ldst + 12) = a3;
#endif
}

// ---------------------------------------------------------------------------
// Kernel 0: zero the output accumulator (float4 vectorized).
// ---------------------------------------------------------------------------
__global__ __launch_bounds__(256) void zero_out_kernel(float4* __restrict__ out, int n4) {
    int i = blockIdx.x * blockDim.x + threadIdx.x;
    if (i < n4) out[i] = make_float4(0.f, 0.f, 0.f, 0.f);
}

// ---------------------------------------------------------------------------
// Kernel 1: h = x @ W^T via V_WMMA_F32_16X16X4_F32 (exact f32 semantics).
// 1250 blocks x 256 threads; each block owns 5 M-tiles (80 rows of x).
// Per wave: all 64 B-fragments of its 16-col W tile are register-resident
// (128 VGPRs, loaded once, reused for 5 tiles = 320 WMMAs); the x-tile is
// double-buffered in LDS with async-to-LDS staging overlapping compute.
// ---------------------------------------------------------------------------
__global__ __launch_bounds__(256) void gemm_wmma_f32_kernel(const float* __restrict__ x,
                                                            const float* __restrict__ w,
                                                            float* __restrict__ h) {
    __shared__ float lx[2][16 * LDS_STRIDE];
    const int tid  = threadIdx.x;
    const int wave = tid >> 5;        // 0..7 -> column tile of W
    const int lane = tid & 31;
    const int half = lane >> 4;       // 0/1: selects K pair within each K4 step
    const int l16  = lane & 15;
    const int mtile0 = blockIdx.x * TILES_PER_BLOCK;

    // Preload all B fragments: B[k][j] = W[j][k]; lane layout mirrors A.
    const float* wrow = w + (size_t)(wave * 16 + l16) * FEAT;
    v2f bfrag[FEAT / 4];
#pragma unroll
    for (int i = 0; i < FEAT / 4; ++i) {
        const int kb = i * 4 + half * 2;
        bfrag[i].x = wrow[kb];
        bfrag[i].y = wrow[kb + 1];
    }

    // Prime the pipeline: stage tile 0 into buffer 0.
    stage_tile(x, mtile0 * 16, &lx[0][0], tid);

    for (int t = 0; t < TILES_PER_BLOCK; ++t) {
        wait_async_lds();          // this wave's async writes to buf[t&1] done
        __syncthreads();           // all waves' writes visible; prior reads done

        if (t + 1 < TILES_PER_BLOCK)
            stage_tile(x, (mtile0 + t + 1) * 16, &lx[(t + 1) & 1][0], tid);

        const float* arow = &lx[t & 1][l16 * LDS_STRIDE];
        v8f c = {};
#pragma unroll
        for (int i = 0; i < FEAT / 4; ++i) {
            const int kb = i * 4 + half * 2;
            v2f a;
            a.x = arow[kb];
            a.y = arow[kb + 1];
            c = __builtin_amdgcn_wmma_f32_16x16x4_f32(
                    /*neg_a=*/false, a, /*neg_b=*/false, bfrag[i],
                    /*c_mod=*/(short)0, c, /*reuse_a=*/false, /*reuse_b=*/false);
        }

        // D layout: VGPR r -> lanes 0-15: M=r, lanes 16-31: M=r+8; N = l16.
        float* hdst = h + (size_t)((mtile0 + t) * 16 + half * 8) * EMBED
                        + wave * 16 + l16;
#pragma unroll
        for (int r = 0; r < 8; ++r) {
            hdst[(size_t)r * EMBED] = c[r];
        }
    }
}

// ---------------------------------------------------------------------------
// Kernel 2: out[row] += val * h[col] over sorted edges.
// One wave owns 32 consecutive edges; lane l owns embed dims [4l, 4l+4).
// Edge metadata is loaded once, coalesced (one edge per lane), then broadcast
// per-iteration with __shfl (uniform index -> v_readlane), so every gather
// address is register-resident up front. Register run-length accumulation
// across equal sorted rows; GLOBAL_ATOMIC_ADD_F32 only at run boundaries.
// h (51.2 MB) is L2-resident -> gather traffic stays on-chip.
// ---------------------------------------------------------------------------
#define EDGES_PER_WAVE 32
#define PREFETCH_DIST  8

__global__ __launch_bounds__(256) void scatter_spmm_kernel(const int*   __restrict__ rows,
                                                           const int*   __restrict__ cols,
                                                           const float* __restrict__ vals,
                                                           const float* __restrict__ h,
                                                           float*       __restrict__ out) {
    const int  gtid = blockIdx.x * blockDim.x + threadIdx.x;
    const long e0   = (long)(gtid >> 5) * EDGES_PER_WAVE;
    if (e0 >= N_EDGES) return;
    const int  lane = threadIdx.x & 31;
    const int  d0   = lane * 4;            // this lane's 4 embed dims

    // Coalesced metadata load: lane i holds edge e0+i.
    const int   r_all = rows[e0 + lane];
    const int   c_all = cols[e0 + lane];
    const float v_all = vals[e0 + lane];

    float4 acc = make_float4(0.f, 0.f, 0.f, 0.f);
    int rprev = __shfl(r_all, 0, 32);

#pragma unroll 4
    for (int i = 0; i < EDGES_PER_WAVE; ++i) {
        const int   r = __shfl(r_all, i, 32);
        const int   c = __shfl(c_all, i, 32);
        const float v = __shfl(v_all, i, 32);

        if (i + PREFETCH_DIST < EDGES_PER_WAVE) {
            const int cp = __shfl(c_all, i + PREFETCH_DIST, 32);
            __builtin_prefetch(h + (size_t)cp * EMBED + d0, 0, 0);
        }

        if (r != rprev) {
            float* dst = out + (size_t)rprev * EMBED + d0;
            unsafeAtomicAdd(dst + 0, acc.x);
            unsafeAtomicAdd(dst + 1, acc.y);
            unsafeAtomicAdd(dst + 2, acc.z);
            unsafeAtomicAdd(dst + 3, acc.w);
            acc = make_float4(0.f, 0.f, 0.f, 0.f);
            rprev = r;
        }
        const float4 hv = *(const float4*)(h + (size_t)c * EMBED + d0);
        acc.x += v * hv.x;
        acc.y += v * hv.y;
        acc.z += v * hv.z;
        acc.w += v * hv.w;
    }
    float* dst = out + (size_t)rprev * EMBED + d0;
    unsafeAtomicAdd(dst + 0, acc.x);
    unsafeAtomicAdd(dst + 1, acc.y);
    unsafeAtomicAdd(dst + 2, acc.z);
    unsafeAtomicAdd(dst + 3, acc.w);
}

// ---------------------------------------------------------------------------
extern "C" void kernel_launch(void* const* d_in, const int* in_sizes, int n_in,
                              void* d_out, int out_size, void* d_ws, size_t ws_size,
                              hipStream_t stream) {
    const float* x     = (const float*)d_in[0];   // [N_NODES, FEAT]
    const float* w     = (const float*)d_in[1];   // [EMBED, FEAT]
    const int*   arow  = (const int*)  d_in[2];   // [N_EDGES] sorted
    const int*   acol  = (const int*)  d_in[3];   // [N_EDGES]
    const float* avals = (const float*)d_in[4];   // [N_EDGES]
    float*       out   = (float*)d_out;           // [N_NODES, EMBED]
    float*       h     = (float*)d_ws;            // [N_NODES, EMBED] scratch (51.2 MB)

    // Phase 0: zero accumulator.
    const int n4 = (N_NODES * EMBED) / 4;
    zero_out_kernel<<<(n4 + 255) / 256, 256, 0, stream>>>((float4*)out, n4);

    // Phase 1: dense GEMM h = x @ W^T (WMMA f32, persistent B, async LDS staging).
    gemm_wmma_f32_kernel<<<N_NODES / (16 * TILES_PER_BLOCK), 256, 0, stream>>>(x, w, h);

    // Phase 2: sorted-edge gather + run-length scatter-add.
    const int waves  = N_EDGES / EDGES_PER_WAVE;      // 100000
    const int blocks = waves / 8;                     // 12500 (8 waves/block)
    scatter_spmm_kernel<<<blocks, 256, 0, stream>>>(arow, acol, avals, h, out);
}